// SpatialTransform_549755813984
// MI455X (gfx1250) — compile-verified
//
#include <hip/hip_runtime.h>

typedef float v2f __attribute__((ext_vector_type(2)));
typedef float v4f __attribute__((ext_vector_type(4)));
typedef float v8f __attribute__((ext_vector_type(8)));

#define N_BATCH 16
#define H 256
#define W 256
#define C 32
#define BPB 32           // reduction blocks per batch image
#define ROWS_PER_BLK 8   // 256 rows / 32 blocks

// ---------------------------------------------------------------------------
// Phase 1: deterministic partial sums over H*W for the global mean pool.
// Each block covers 8 contiguous rows (2048 pixels) of one image.
// Thread (s,q): s = pixel slot (0..31), q = float4 channel group (0..7).
// Loads are 512B-contiguous per wave per iteration; x stays RT so it lands
// (and stays) in the 192MB L2 for the phase-3 gather.
// ---------------------------------------------------------------------------
__global__ void __launch_bounds__(256) st_reduce_partial(
    const float* __restrict__ x, float* __restrict__ partial) {
  __shared__ v4f sdata[256];
  const int blk = blockIdx.x;       // 0 .. N_BATCH*BPB-1
  const int n = blk / BPB;
  const int b = blk % BPB;
  const int t = threadIdx.x;
  const int q = t & 7;              // float4 group within 32 channels
  const int s = t >> 3;             // pixel slot 0..31

  const v4f* px = (const v4f*)(
      x + ((size_t)n * H * W + (size_t)b * ROWS_PER_BLK * W) * C);

  v4f acc = {0.f, 0.f, 0.f, 0.f};
#pragma unroll 4
  for (int i = 0; i < (ROWS_PER_BLK * W) / 32; ++i) {
    const int p = s + 32 * i;
    acc += px[(size_t)p * (C / 4) + q];
  }
  sdata[t] = acc;
  __syncthreads();

  if (t < 8) {  // one thread per channel group sums the 32 slots
    v4f sum = {0.f, 0.f, 0.f, 0.f};
#pragma unroll
    for (int ss = 0; ss < 32; ++ss) sum += sdata[ss * 8 + t];
    ((v4f*)(partial + (size_t)blk * C))[t] = sum;
  }
}

// ---------------------------------------------------------------------------
// Phase 2: loc network via V_WMMA_F32_16X16X4_F32.
// mean (16x32) @ W_loc (32x6, zero-padded to 32x16) accumulated over 8
// K=4 steps; D = 16x16 f32, columns 0..5 are the affine params; + b_loc.
// Single wave32, EXEC all-1s through the WMMA chain.
// A layout (16x4): lanes 0-15 hold M=lane, lanes 16-31 hold M=lane-16;
//   vgpr0 = K0(lo-half)/K2(hi-half), vgpr1 = K1/K3.
// B layout (4x16): lanes hold N=lane&15, same K striping as A.
// D layout: vgpr j -> M = j + 8*(lane>=16), N = lane&15.
// ---------------------------------------------------------------------------
__global__ void __launch_bounds__(32) st_loc_theta(
    const float* __restrict__ partial, const float* __restrict__ W_loc,
    const float* __restrict__ b_loc, float* __restrict__ theta) {
  __shared__ float lmean[N_BATCH * C];
  const int lane = threadIdx.x;     // == channel for the mean stage

  // Finish the mean: 32 partials per (image, channel).
  for (int M = 0; M < N_BATCH; ++M) {
    float sum = 0.f;
#pragma unroll 8
    for (int b = 0; b < BPB; ++b)
      sum += partial[(size_t)(M * BPB + b) * C + lane];
    lmean[M * C + lane] = sum * (1.0f / (float)(H * W));
  }
  __syncthreads();

  const int row = lane & 15;        // M for A, N for B/D
  const int hi = lane >> 4;
  const int ncol = (row < 6) ? row : 5;  // clamped index: unpredicated loads
  const bool valid = (row < 6);

  v8f acc = {};
#pragma unroll
  for (int k = 0; k < 8; ++k) {
    const int k0 = 4 * k + 2 * hi;
    v2f a, bm;
    a.x = lmean[row * C + k0];
    a.y = lmean[row * C + k0 + 1];
    const float w0 = W_loc[k0 * 6 + ncol];
    const float w1 = W_loc[(k0 + 1) * 6 + ncol];
    bm.x = valid ? w0 : 0.0f;
    bm.y = valid ? w1 : 0.0f;
    acc = __builtin_amdgcn_wmma_f32_16x16x4_f32(
        /*neg_a=*/false, a, /*neg_b=*/false, bm,
        /*c_mod=*/(short)0, acc, /*reuse_a=*/false, /*reuse_b=*/false);
  }

  if (valid) {
    const float bb = b_loc[row];
#pragma unroll
    for (int j = 0; j < 8; ++j) {
      const int M = j + 8 * hi;
      theta[M * 6 + row] = acc[j] + bb;
    }
  }
}

// ---------------------------------------------------------------------------
// Phase 3: affine grid + bilinear gather. 8 lanes per output pixel, each
// handling a float4 of channels (32 channels = 128B coalesced per pixel).
// Gathers hit L2 (x resident from phase 1); output stores are non-temporal
// so 134MB of writes don't evict x.
// ---------------------------------------------------------------------------
__global__ void __launch_bounds__(256) st_sampler(
    const float* __restrict__ x, const float* __restrict__ theta,
    float* __restrict__ out) {
  const int t = threadIdx.x;
  const int q = t & 7;                          // float4 channel group
  const int pix = blockIdx.x * 32 + (t >> 3);   // global pixel id
  const int n = pix >> 16;
  const int oy = (pix >> 8) & 255;
  const int ox = pix & 255;

  const float* th = theta + n * 6;
  const float gy = -1.0f + (float)oy * (2.0f / 255.0f);
  const float gx = -1.0f + (float)ox * (2.0f / 255.0f);
  const float uy = (th[0] * gy + th[1] * gx + th[2] + 1.0f) * (0.5f * (H - 1));
  const float ux = (th[3] * gy + th[4] * gx + th[5] + 1.0f) * (0.5f * (W - 1));

  const float fy = floorf(uy), fx = floorf(ux);
  const float ry = uy - fy, rx = ux - fx;
  const int iy = (int)fy, ix = (int)fx;
  const int y0 = min(max(iy, 0), H - 1);
  const int y1 = min(max(iy + 1, 0), H - 1);
  const int x0 = min(max(ix, 0), W - 1);
  const int x1 = min(max(ix + 1, 0), W - 1);

  const float w00 = (1.f - ry) * (1.f - rx);
  const float w01 = (1.f - ry) * rx;
  const float w10 = ry * (1.f - rx);
  const float w11 = ry * rx;

  const v4f* xb = (const v4f*)(x + (size_t)n * H * W * C);
  const v4f c00 = xb[(size_t)(y0 * W + x0) * (C / 4) + q];
  const v4f c01 = xb[(size_t)(y0 * W + x1) * (C / 4) + q];
  const v4f c10 = xb[(size_t)(y1 * W + x0) * (C / 4) + q];
  const v4f c11 = xb[(size_t)(y1 * W + x1) * (C / 4) + q];

  const v4f r = w00 * c00 + w01 * c01 + w10 * c10 + w11 * c11;

  v4f* op = (v4f*)(out + (size_t)pix * C) + q;
  __builtin_nontemporal_store(r, op);  // NT store: keep x resident in L2
}

// ---------------------------------------------------------------------------
extern "C" void kernel_launch(void* const* d_in, const int* in_sizes, int n_in,
                              void* d_out, int out_size, void* d_ws,
                              size_t ws_size, hipStream_t stream) {
  const float* x = (const float*)d_in[0];      // (16,256,256,32) f32
  const float* W_loc = (const float*)d_in[1];  // (32,6) f32
  const float* b_loc = (const float*)d_in[2];  // (6,) f32
  float* out = (float*)d_out;                  // (16,256,256,32) f32

  float* partial = (float*)d_ws;               // N_BATCH*BPB*C = 16384 f32
  float* theta = partial + N_BATCH * BPB * C;  // 16*6 = 96 f32

  st_reduce_partial<<<N_BATCH * BPB, 256, 0, stream>>>(x, partial);
  st_loc_theta<<<1, 32, 0, stream>>>(partial, W_loc, b_loc, theta);
  st_sampler<<<(N_BATCH * H * W) / 32, 256, 0, stream>>>(x, theta, out);
}